// MultiHeadedAttention_18537078849752
// MI455X (gfx1250) — compile-verified
//
#include <hip/hip_runtime.h>

typedef float v2f __attribute__((ext_vector_type(2)));
typedef float v8f __attribute__((ext_vector_type(8)));

#define NH 8
#define BSZ 2
#define LSEQ 256
#define DMODEL 512
#define DK 64
#define BHN (BSZ*NH)
#define LDSS 260  // padded LDS row stride (floats) to avoid bank conflicts

// ---------------------------------------------------------------------------
// Fragment loader for V_WMMA_F32_16X16X4_F32.
// A (16x4, MxK): lanes 0-15 hold M=lane, VGPR0/1 = K {0,1}; lanes 16-31 = K {2,3}.
// B (4x16, KxN) from an N-major source S[n][k]: identical gather with n as row.
// Source must be row-major with K contiguous (stride ld).
// ---------------------------------------------------------------------------
__device__ __forceinline__ v2f load_frag(const float* __restrict__ base, int ld,
                                         int r0, int k) {
  int lane = threadIdx.x & 31;
  int r  = r0 + (lane & 15);
  int kk = k + ((lane >> 4) << 1);
  const float* p = base + (size_t)r * ld + kk;
  v2f out;
  out.x = p[0];
  out.y = p[1];
  return out;
}

__device__ __forceinline__ v8f wmma_f32(v2f a, v2f b, v8f c) {
  return __builtin_amdgcn_wmma_f32_16x16x4_f32(false, a, false, b, (short)0, c,
                                               false, false);
}

// ---------------------------------------------------------------------------
// Kernel 1: fused Q/K/V projections. y = X @ W^T + b, stored head-split
// [BH, L, dk]. One 16x16 output tile per wave; 3 projs * 32*32 tiles = 3072
// waves = 384 blocks of 256 threads.
// ---------------------------------------------------------------------------
__global__ __launch_bounds__(256) void qkv_proj_kernel(
    const float* __restrict__ query, const float* __restrict__ keyin,
    const float* __restrict__ value,
    const float* __restrict__ Wq, const float* __restrict__ bq,
    const float* __restrict__ Wk, const float* __restrict__ bk,
    const float* __restrict__ Wv, const float* __restrict__ bv,
    float* __restrict__ qh, float* __restrict__ kh, float* __restrict__ vh) {
  int wave = (blockIdx.x * blockDim.x + threadIdx.x) >> 5;
  int proj = wave / (32 * 32);
  int t    = wave % (32 * 32);
  int m0 = (t >> 5) << 4;
  int n0 = (t & 31) << 4;

  const float* X    = (proj == 0) ? query : (proj == 1) ? keyin : value;
  const float* W    = (proj == 0) ? Wq    : (proj == 1) ? Wk    : Wv;
  const float* bias = (proj == 0) ? bq    : (proj == 1) ? bk    : bv;
  float*       out  = (proj == 0) ? qh    : (proj == 1) ? kh    : vh;

  v8f c = {};
  for (int k = 0; k < DMODEL; k += 4) {
    v2f a = load_frag(X, DMODEL, m0, k);
    v2f b = load_frag(W, DMODEL, n0, k);
    c = wmma_f32(a, b, c);
  }

  int lane = threadIdx.x & 31;
  int n = n0 + (lane & 15);
  float bn = bias[n];
  int h = n >> 6, d = n & 63;
#pragma unroll
  for (int vg = 0; vg < 8; ++vg) {
    int m = m0 + vg + ((lane >> 4) << 3);  // C/D: lanes 16-31 hold M=vgpr+8
    int b_ = m >> 8, l = m & 255;
    out[(((size_t)(b_ * NH + h)) * LSEQ + l) * DK + d] = c[vg] + bn;
  }
}

// ---------------------------------------------------------------------------
// Kernel 2: fused scores + relative_q bias + mask + softmax + (V + relative_v)
// context. One block (8 waves) per (bh, 16-row tile of i).
// ---------------------------------------------------------------------------
__global__ __launch_bounds__(256) void attn_kernel(
    const float* __restrict__ qh, const float* __restrict__ kh,
    const float* __restrict__ vh, const int* __restrict__ mask,
    const float* __restrict__ relq, const float* __restrict__ relv,
    float* __restrict__ ctxo) {
  __shared__ float S[16 * LDSS];

  int bh = blockIdx.y;
  int i0 = blockIdx.x << 4;
  int b  = bh / NH;
  int hh = bh % NH;
  int tid  = threadIdx.x;
  int lane = tid & 31;
  int w    = tid >> 5;  // wave id 0..7

  const float* qb = qh + (size_t)bh * LSEQ * DK;
  const float* kb = kh + (size_t)bh * LSEQ * DK;
  const float* vb = vh + (size_t)bh * LSEQ * DK;

  // ---- Phase A: raw q.k^T scores via WMMA, 16x256 tile into LDS ----
#pragma unroll
  for (int jt = w * 2; jt < w * 2 + 2; ++jt) {
    int j0 = jt << 4;
    v8f c = {};
#pragma unroll
    for (int k = 0; k < DK; k += 4) {
      v2f a = load_frag(qb, DK, i0, k);
      v2f bf = load_frag(kb, DK, j0, k);
      c = wmma_f32(a, bf, c);
    }
    int jl = lane & 15;
#pragma unroll
    for (int vg = 0; vg < 8; ++vg) {
      int il = vg + ((lane >> 4) << 3);
      S[il * LDSS + j0 + jl] = c[vg];
    }
  }
  __syncthreads();

  // ---- Phase A2: add relative_q bias, scale, mask. thread = column j ----
  const float scale = 0.125f;  // 1/sqrt(64)
  for (int il = 0; il < 16; ++il) {
    int i = i0 + il;
    const float* qrow = qb + (size_t)i * DK;
    const float* rrow = relq + (((size_t)bh * LSEQ + i) * LSEQ + tid) * DK;
    if (il < 15) __builtin_prefetch(rrow + (size_t)LSEQ * DK, 0, 1);
    float acc = 0.f;
#pragma unroll
    for (int t = 0; t < DK; t += 4) {
      float4 qv = *(const float4*)(qrow + t);
      float4 rv = *(const float4*)(rrow + t);
      acc += qv.x * rv.x + qv.y * rv.y + qv.z * rv.z + qv.w * rv.w;
    }
    float s = S[il * LDSS + tid] * scale + acc;
    if (mask[((size_t)b * LSEQ + i) * LSEQ + tid] == 0) s = -1e9f;
    S[il * LDSS + tid] = s;
  }
  __syncthreads();

  // ---- Phase B: row softmax (wave32 shuffle reductions), 2 rows per wave ----
#pragma unroll
  for (int il = w * 2; il < w * 2 + 2; ++il) {
    float vals[8];
    float m = -3.4e38f;
#pragma unroll
    for (int t = 0; t < 8; ++t) {
      vals[t] = S[il * LDSS + t * 32 + lane];
      m = fmaxf(m, vals[t]);
    }
    for (int off = 16; off; off >>= 1) m = fmaxf(m, __shfl_xor(m, off, 32));
    float sum = 0.f;
#pragma unroll
    for (int t = 0; t < 8; ++t) {
      vals[t] = expf(vals[t] - m);
      sum += vals[t];
    }
    for (int off = 16; off; off >>= 1) sum += __shfl_xor(sum, off, 32);
    float inv = 1.f / sum;
#pragma unroll
    for (int t = 0; t < 8; ++t) S[il * LDSS + t * 32 + lane] = vals[t] * inv;
  }
  __syncthreads();

  // ---- Phase C: ctx[i,d] = sum_j p[i,j] * (v[j,d] + relv[i,j,d]) ----
#pragma unroll
  for (int il = w * 2; il < w * 2 + 2; ++il) {
    int i = i0 + il;
    const float* rbase = relv + (((size_t)bh * LSEQ + i) * LSEQ) * DK;
    float ax = 0.f, ay = 0.f;
    for (int j = 0; j < LSEQ; ++j) {
      float p = S[il * LDSS + j];  // broadcast read
      float2 vv = *(const float2*)(vb + (size_t)j * DK + lane * 2);
      float2 rv = *(const float2*)(rbase + (size_t)j * DK + lane * 2);
      ax = fmaf(p, vv.x + rv.x, ax);
      ay = fmaf(p, vv.y + rv.y, ay);
    }
    // store in [B, L, D] layout for the output projection
    float* o = ctxo + ((size_t)b * LSEQ + i) * DMODEL + hh * DK + lane * 2;
    o[0] = ax;
    o[1] = ay;
  }
}

// ---------------------------------------------------------------------------
// Kernel 3: output projection out = ctx @ Wo^T + bo. 1024 waves = 128 blocks.
// ---------------------------------------------------------------------------
__global__ __launch_bounds__(256) void out_proj_kernel(
    const float* __restrict__ ctxo, const float* __restrict__ Wo,
    const float* __restrict__ bo, float* __restrict__ out) {
  int wave = (blockIdx.x * blockDim.x + threadIdx.x) >> 5;
  int m0 = (wave >> 5) << 4;
  int n0 = (wave & 31) << 4;

  v8f c = {};
  for (int k = 0; k < DMODEL; k += 4) {
    v2f a = load_frag(ctxo, DMODEL, m0, k);
    v2f b = load_frag(Wo, DMODEL, n0, k);
    c = wmma_f32(a, b, c);
  }
  int lane = threadIdx.x & 31;
  int n = n0 + (lane & 15);
  float bn = bo[n];
#pragma unroll
  for (int vg = 0; vg < 8; ++vg) {
    int m = m0 + vg + ((lane >> 4) << 3);
    out[(size_t)m * DMODEL + n] = c[vg] + bn;
  }
}

// ---------------------------------------------------------------------------
extern "C" void kernel_launch(void* const* d_in, const int* in_sizes, int n_in,
                              void* d_out, int out_size, void* d_ws,
                              size_t ws_size, hipStream_t stream) {
  const float* query = (const float*)d_in[0];
  const float* key   = (const float*)d_in[1];
  const float* value = (const float*)d_in[2];
  const int*   mask  = (const int*)d_in[3];
  const float* relq  = (const float*)d_in[4];
  const float* relv  = (const float*)d_in[5];
  const float* Wq = (const float*)d_in[6];
  const float* bq = (const float*)d_in[7];
  const float* Wk = (const float*)d_in[8];
  const float* bk = (const float*)d_in[9];
  const float* Wv = (const float*)d_in[10];
  const float* bv = (const float*)d_in[11];
  const float* Wo = (const float*)d_in[12];
  const float* bo = (const float*)d_in[13];
  float* out = (float*)d_out;

  // workspace layout: qh | kh | vh (each [BH,L,dk]) | ctx ([B,L,D]) = 4 MB
  const size_t seg = (size_t)BHN * LSEQ * DK;  // 262144 floats = 1 MB
  float* qh  = (float*)d_ws;
  float* kh  = qh + seg;
  float* vh  = kh + seg;
  float* ctx = vh + seg;

  qkv_proj_kernel<<<384, 256, 0, stream>>>(query, key, value, Wq, bq, Wk, bk,
                                           Wv, bv, qh, kh, vh);
  attn_kernel<<<dim3(LSEQ / 16, BHN), 256, 0, stream>>>(qh, kh, vh, mask, relq,
                                                        relv, ctx);
  out_proj_kernel<<<128, 256, 0, stream>>>(ctx, Wo, bo, out);
}